// CAE_88381837017318
// MI455X (gfx1250) — compile-verified
//
#include <hip/hip_runtime.h>
#include <hip/hip_bf16.h>
#include <math.h>

typedef __attribute__((ext_vector_type(2))) float v2f;
typedef __attribute__((ext_vector_type(8))) float v8f;

// D = A(16x4) * B(4x16) + C, fp32 WMMA (V_WMMA_F32_16X16X4_F32).
__device__ __forceinline__ v8f wmma4(v2f a, v2f b, v8f c) {
    return __builtin_amdgcn_wmma_f32_16x16x4_f32(false, a, false, b, (short)0, c,
                                                 false, false);
}

// ---------------------------------------------------------------------------
// Zero fill
// ---------------------------------------------------------------------------
__global__ void zerof_k(float* __restrict__ p, int n) {
    const int i = blockIdx.x * blockDim.x + threadIdx.x;
    if (i < n) p[i] = 0.f;
}

// ---------------------------------------------------------------------------
// Pad x: [64,120,64,64] -> [64,120,66,66] with zero border (one stream pass)
// ---------------------------------------------------------------------------
__global__ void pad_x_k(const float* __restrict__ x, float* __restrict__ xp) {
    const int idx = blockIdx.x * blockDim.x + threadIdx.x;
    if (idx >= 64 * 120 * 66 * 66) return;
    const int xq = idx % 66;
    const int yq = (idx / 66) % 66;
    const int c  = (idx / 4356) % 120;
    const int b  = idx / (4356 * 120);
    float v = 0.f;
    if (xq >= 1 && xq <= 64 && yq >= 1 && yq <= 64)
        v = x[(((size_t)b * 120 + c) * 64 + (yq - 1)) * 64 + (xq - 1)];
    xp[idx] = v;
}

// ---------------------------------------------------------------------------
// Weight pre-pack: w[oc][ic][ky][kx] -> wpk[tap][ic/2][oc][2]
// A-fragment (K,K+1) pair becomes ONE coalesced b64 load.
// ---------------------------------------------------------------------------
__global__ void pack_w_k(const float* __restrict__ w, float* __restrict__ wpk,
                         int CI, int CO) {
    const int i = blockIdx.x * blockDim.x + threadIdx.x;
    const int total = CO * CI * 9;
    if (i >= total) return;
    const int tap = i % 9;
    const int ic = (i / 9) % CI;
    const int oc = i / (9 * CI);
    wpk[(((size_t)tap * (CI >> 1) + (ic >> 1)) * CO + oc) * 2 + (ic & 1)] = w[i];
}

// ---------------------------------------------------------------------------
// Generic conv(3x3) [+relu] + maxpool2 as WMMA implicit GEMM.
// Input is spatially zero-padded [CI][HI+2][HI+2] -> NO masks, NO clamps:
// inner loop is pure loads feeding WMMA (1 b64 A-load + 4 b32 B-loads / 2 WMMA).
// One wave = 16 oc x 16 px tile for TWO adjacent rows (the pool pair).
// Output: padded [CO][HO+2][HO+2] (border pre-zeroed) or unpadded [CO][HO][HO].
// ---------------------------------------------------------------------------
template <int CI, int CO, int HI, bool RELU, bool PADOUT>
__global__ void conv_pool_wmma_k(const float* __restrict__ in,
                                 const float* __restrict__ wpk,
                                 const float* __restrict__ bias,
                                 float* __restrict__ out) {
    constexpr int HO = HI / 2;
    constexpr int HIP = HI + 2;
    constexpr int HP2 = HIP * HIP;
    constexpr int HOP = PADOUT ? (HO + 2) : HO;
    constexpr int OOF = PADOUT ? 1 : 0;          // output border offset
    constexpr int XT = (HI + 15) / 16;           // 16-px tiles per row
    constexpr int OCT = CO / 16;                 // 16-oc tiles

    const int lane = threadIdx.x & 31;
    int t = blockIdx.x;
    const int xt = t % XT;  t /= XT;
    const int ot = t % OCT; t /= OCT;
    const int y2 = t % HO;  t /= HO;
    const int b  = t;

    const int x0 = xt << 4;
    const int n = lane & 15;
    const int half = lane >> 4;
    const int y0 = y2 * 2;

    v8f acc0 = {0.f,0.f,0.f,0.f,0.f,0.f,0.f,0.f};
    v8f acc1 = {0.f,0.f,0.f,0.f,0.f,0.f,0.f,0.f};
    const float* inb = in + (size_t)b * CI * HP2;

    #pragma unroll
    for (int ky = 0; ky < 3; ++ky) {
        #pragma unroll
        for (int kx = 0; kx < 3; ++kx) {
            const int tap = ky * 3 + kx;
            // A pairs: [tap][k/2][oc][2] as v2f; lane pair index = k0/2 + half
            const v2f* ap = (const v2f*)wpk
                          + ((size_t)tap * (CI / 2) + half) * CO + ot * 16 + n;
            // B: padded rows, always in-bounds
            const float* p0 = inb + (size_t)(half * 2) * HP2
                                  + (size_t)(y0 + ky) * HIP + (x0 + n + kx);
            const float* p1 = p0 + HIP;
            #pragma unroll 2
            for (int kc = 0; kc < CI / 4; ++kc) {
                v2f af = *ap;
                v2f bf0, bf1;
                bf0.x = p0[0];
                bf0.y = p0[HP2];
                bf1.x = p1[0];
                bf1.y = p1[HP2];
                acc0 = wmma4(af, bf0, acc0);
                acc1 = wmma4(af, bf1, acc1);
                ap += 2 * CO;
                p0 += 4 * HP2;
                p1 += 4 * HP2;
            }
        }
    }

    // bias [+relu] + 2x2 maxpool (rows via two accumulators, cols via lane-xor)
    #pragma unroll
    for (int r = 0; r < 8; ++r) {
        const int oc = ot * 16 + r + half * 8;
        const float bb = bias[oc];
        float v0 = acc0[r] + bb;
        float v1 = acc1[r] + bb;
        if (RELU) { v0 = fmaxf(v0, 0.f); v1 = fmaxf(v1, 0.f); }
        float mv = fmaxf(v0, v1);
        float ov = fmaxf(mv, __shfl_xor(mv, 1, 32));
        const int xo = x0 + n;
        if ((n & 1) == 0 && xo < HI) {
            out[(((size_t)b * CO + oc) * HOP + (y2 + OOF)) * HOP + ((xo >> 1) + OOF)] = ov;
        }
    }
}

// ---------------------------------------------------------------------------
// FC layer as WMMA GEMM: out[M,N] = [relu](A[M,K] * W[N,K]^T + bias), M=64.
// One wave per 16x16 output tile. 4 waves / block.
// ---------------------------------------------------------------------------
template <bool RELU>
__global__ void fc_wmma_k(const float* __restrict__ A,
                          const float* __restrict__ W,
                          const float* __restrict__ bias,
                          float* __restrict__ out, int N, int K) {
    const int lane = threadIdx.x & 31;
    const int wave = threadIdx.x >> 5;
    const int tile = blockIdx.x * 4 + wave;
    const int tn = N >> 4;
    const int m0 = (tile / tn) << 4;
    const int n0 = (tile % tn) << 4;
    const int q = lane & 15;
    const int half = lane >> 4;

    v8f acc = {0.f,0.f,0.f,0.f,0.f,0.f,0.f,0.f};
    const float* arow = A + (size_t)(m0 + q) * K + half * 2;
    const float* wrow = W + (size_t)(n0 + q) * K + half * 2;
    #pragma unroll 4
    for (int k0 = 0; k0 < K; k0 += 4) {
        v2f af = *(const v2f*)(arow + k0);
        v2f bf = *(const v2f*)(wrow + k0);
        acc = wmma4(af, bf, acc);
    }
    #pragma unroll
    for (int r = 0; r < 8; ++r) {
        const int m = m0 + r + half * 8;
        const int nn = n0 + q;
        float vv = acc[r] + bias[nn];
        if (RELU) vv = fmaxf(vv, 0.f);
        out[(size_t)m * N + nn] = vv;
    }
}

// FC3 (256 -> 6) + hardtanh
__global__ void fc3_hardtanh_k(const float* __restrict__ h2,
                               const float* __restrict__ wl3,
                               const float* __restrict__ bl3,
                               float* __restrict__ params) {
    const int t = blockIdx.x * blockDim.x + threadIdx.x;
    if (t >= 64 * 6) return;
    const int m = t / 6, n = t % 6;
    float s = bl3[n];
    const float* hr = h2 + m * 256;
    const float* wr = wl3 + n * 256;
    for (int k = 0; k < 256; ++k) s += hr[k] * wr[k];
    params[t] = fminf(fmaxf(s, -1.f), 1.f);
}

__global__ void init_max_k(unsigned* __restrict__ maxbuf) {
    if (threadIdx.x < 64) maxbuf[threadIdx.x] = 0u;
}

__device__ __forceinline__ float dereg(float t, float lo, float hi) {
    return (t + 1.f) * (hi - lo) * 0.5f + lo;
}

// Per-pixel fields: sb, vel, bin; v output; per-batch max(sb) for normalizer
__global__ void fields_k(const float* __restrict__ params,
                         const float* __restrict__ xxg,
                         const float* __restrict__ yyg,
                         float* __restrict__ vout,
                         float* __restrict__ sbuf,
                         int* __restrict__ binbuf,
                         unsigned* __restrict__ maxbuf) {
    const int b = blockIdx.x;
    const float* p = params + b * 6;
    const float pos = atan2f(p[0], p[1]);
    const float inc = dereg(p[2], 5.f, 90.f);   // used directly in cos/sin, as in ref
    const float a   = dereg(p[3], 0.1f, 0.4f);
    const float ah  = dereg(p[4], 0.1f, 1.0f);
    const float Vh  = dereg(p[5], 50.f, 500.f);
    const float cp = cosf(pos), sp = sinf(pos);
    const float ci = cosf(inc), si = sinf(inc);
    const float PI = 3.14159265358979323846f;

    for (int pix = threadIdx.x; pix < 4096; pix += blockDim.x) {
        const float xv = xxg[pix], yv = yyg[pix];
        const float xt = xv * cp + yv * sp;
        const float yt = -xv * sp + yv * ci * cp;
        const float rr = sqrtf(xt * xt + yt * yt);
        const float sb = expf(-rr / a);
        float vel = sqrtf(Vh * Vh * (1.f - (ah / rr) * atanf(rr / ah)));
        vel = vel * (-cosf(PI - atan2f(xt, yt) + pos) * si);
        vout[b * 4096 + pix] = vel;
        float binf = floorf(vel / 10.0f) + 60.f;
        binf = fminf(fmaxf(binf, 0.f), 120.f);
        // NaN or ==120 -> no bin written (matches reference one-hot semantics)
        const int bin = (binf >= 0.f && binf <= 119.f) ? (int)binf : 120;
        sbuf[b * 4096 + pix] = sb;
        binbuf[b * 4096 + pix] = bin;
        if (bin < 120) atomicMax(maxbuf + b, __float_as_uint(sb));  // sb > 0
    }
}

// Output-linear cube fill: out[b][vb][pix] = (bin==vb) ? sb/max : 0
__global__ void cube_k(const float* __restrict__ sbuf,
                       const int* __restrict__ binbuf,
                       const unsigned* __restrict__ maxbuf,
                       float* __restrict__ cube) {
    const int pix = blockIdx.x * blockDim.x + threadIdx.x;  // 16*256 = 4096
    const int vb = blockIdx.y;
    const int b = blockIdx.z;
    const float inv = 1.f / __uint_as_float(maxbuf[b]);
    const int bin = binbuf[b * 4096 + pix];
    const float val = (bin == vb) ? sbuf[b * 4096 + pix] * inv : 0.f;
    cube[(((size_t)b * 120 + vb) << 12) + pix] = val;
}

// ---------------------------------------------------------------------------
extern "C" void kernel_launch(void* const* d_in, const int* in_sizes, int n_in,
                              void* d_out, int out_size, void* d_ws, size_t ws_size,
                              hipStream_t stream) {
    const float* x   = (const float*)d_in[0];
    const float* w0  = (const float*)d_in[1];
    const float* b0  = (const float*)d_in[2];
    const float* w1  = (const float*)d_in[3];
    const float* b1  = (const float*)d_in[4];
    const float* w2  = (const float*)d_in[5];
    const float* b2  = (const float*)d_in[6];
    const float* w3  = (const float*)d_in[7];
    const float* b3  = (const float*)d_in[8];
    const float* wl1 = (const float*)d_in[9];
    const float* bl1 = (const float*)d_in[10];
    const float* wl2 = (const float*)d_in[11];
    const float* bl2 = (const float*)d_in[12];
    const float* wl3 = (const float*)d_in[13];
    const float* bl3 = (const float*)d_in[14];
    const float* xxg = (const float*)d_in[15];
    const float* yyg = (const float*)d_in[16];

    float* out = (float*)d_out;                 // cube [64,120,64,64] then v [64,64,64]
    float* vout = out + (size_t)64 * 120 * 4096;

    float* ws = (float*)d_ws;
    float* xp  = ws;                             // 64*120*66*66 = 33,454,080
    float* a0p = xp + 33454080;                  // 64*16*34*34  = 1,183,744
    float* a1p = a0p + 1183744;                  // 64*32*18*18  = 663,552
    float* a2p = a1p + 663552;                   // 64*64*10*10  = 409,600
    float* a3  = a2p + 409600;                   // 64*128*4*4   = 131,072 (flat [64,2048])
    float* h1  = a3 + 131072;                    // 64*1024
    float* h2  = h1 + 65536;                     // 64*256
    float* pr  = h2 + 16384;                     // 64*6
    float* sbuf = pr + 384;                      // 64*4096
    int*   binbuf = (int*)(sbuf + 262144);       // 64*4096 ints
    unsigned* maxbuf = (unsigned*)(binbuf + 262144); // 64
    float* wpk0 = (float*)(maxbuf + 64);         // 9*120*16 = 17280
    float* wpk1 = wpk0 + 17280;                  // 9*16*32  = 4608
    float* wpk2 = wpk1 + 4608;                   // 9*32*64  = 18432
    float* wpk3 = wpk2 + 18432;                  // 9*64*128 = 73728

    // input padding + weight packing + padded-activation border zeroing
    pad_x_k<<<dim3((64*120*66*66 + 255) / 256), dim3(256), 0, stream>>>(x, xp);
    pack_w_k<<<dim3((17280 + 255) / 256), dim3(256), 0, stream>>>(w0, wpk0, 120, 16);
    pack_w_k<<<dim3((4608 + 255) / 256),  dim3(256), 0, stream>>>(w1, wpk1, 16, 32);
    pack_w_k<<<dim3((18432 + 255) / 256), dim3(256), 0, stream>>>(w2, wpk2, 32, 64);
    pack_w_k<<<dim3((73728 + 255) / 256), dim3(256), 0, stream>>>(w3, wpk3, 64, 128);
    zerof_k<<<dim3((1183744 + 255) / 256), dim3(256), 0, stream>>>(a0p, 1183744);
    zerof_k<<<dim3((663552 + 255) / 256),  dim3(256), 0, stream>>>(a1p, 663552);
    zerof_k<<<dim3((409600 + 255) / 256),  dim3(256), 0, stream>>>(a2p, 409600);

    // conv encoder (all WMMA implicit GEMM on padded inputs; one wave per block)
    conv_pool_wmma_k<120, 16, 64, false, true ><<<dim3(64 * 32 * 1 * 4), dim3(32), 0, stream>>>(xp,  wpk0, b0, a0p);
    conv_pool_wmma_k<16,  32, 32, true,  true ><<<dim3(64 * 16 * 2 * 2), dim3(32), 0, stream>>>(a0p, wpk1, b1, a1p);
    conv_pool_wmma_k<32,  64, 16, true,  true ><<<dim3(64 * 8 * 4 * 1),  dim3(32), 0, stream>>>(a1p, wpk2, b2, a2p);
    conv_pool_wmma_k<64, 128, 8,  true,  false><<<dim3(64 * 4 * 8 * 1),  dim3(32), 0, stream>>>(a2p, wpk3, b3, a3);

    // linear encoder (WMMA GEMMs), then 256->6 + hardtanh
    fc_wmma_k<true><<<dim3(64), dim3(128), 0, stream>>>(a3, wl1, bl1, h1, 1024, 2048);
    fc_wmma_k<true><<<dim3(16), dim3(128), 0, stream>>>(h1, wl2, bl2, h2, 256, 1024);
    fc3_hardtanh_k<<<dim3(2), dim3(256), 0, stream>>>(h2, wl3, bl3, pr);

    // cube maker
    init_max_k<<<dim3(1), dim3(64), 0, stream>>>(maxbuf);
    fields_k<<<dim3(64), dim3(256), 0, stream>>>(pr, xxg, yyg, vout, sbuf, binbuf, maxbuf);
    cube_k<<<dim3(16, 120, 64), dim3(256), 0, stream>>>(sbuf, binbuf, maxbuf, out);
}